// SCENE15_JointMembership_89816356094167
// MI455X (gfx1250) — compile-verified
//
#include <hip/hip_runtime.h>
#include <cstdint>

// ---------------------------------------------------------------------------
// SCENE15: 2-qubit circuit expectation values, closed form:
//   z0 = cos(t_ry)*cos(x0) - sin(t_ry)*cos(t_rz)*sin(x0)*sin(x1)
//   z1 = cos(t_ry)*cos(x1)
//   out = (z+1)*0.5
// Pure streaming map: 8B in -> 8B out per batch element; HBM-bound (~2.9us
// at 23.3 TB/s for B=4.19M). Data path: gfx1250 async global->LDS copies,
// double-buffered per wave, gated by s_wait_asynccnt.
// ---------------------------------------------------------------------------

#define BLOCK 256
#define ITERS 4

#if defined(__has_builtin)
#  if __has_builtin(__builtin_amdgcn_global_load_async_to_lds_b128) && \
      __has_builtin(__builtin_amdgcn_s_wait_asynccnt)
#    define USE_ASYNC_LDS 1
#  else
#    define USE_ASYNC_LDS 0
#  endif
#else
#  define USE_ASYNC_LDS 0
#endif

// Raw 4x32-bit vector type matching the builtin's expected pointee
// ("int __attribute__((__vector_size__(4 * sizeof(int))))" per clang's
// diagnostic), in the global (1) and LDS (3) address spaces.
typedef int v4i __attribute__((vector_size(16)));
typedef __attribute__((address_space(1))) v4i g_v4i;
typedef __attribute__((address_space(3))) v4i l_v4i;

__device__ __forceinline__ void qmap(float x0, float x1, float A, float Bc,
                                     float& o0, float& o1) {
    // Native trig: v_sin_f32 / v_cos_f32 (+ 1/(2pi) prescale) — 2 VALU each.
    float s0 = __sinf(x0);
    float c0 = __cosf(x0);
    float s1 = __sinf(x1);
    float c1 = __cosf(x1);
    float z0 = fmaf(-Bc * s0, s1, A * c0);
    float z1 = A * c1;
    o0 = fmaf(z0, 0.5f, 0.5f);
    o1 = fmaf(z1, 0.5f, 0.5f);
}

__global__ void __launch_bounds__(BLOCK)
qsim2_stream_kernel(const float4* __restrict__ x4,
                    const float*  __restrict__ th_rz,
                    const float*  __restrict__ th_ry,
                    float4* __restrict__ out4,
                    int n4) {
    // Uniform circuit parameters (broadcast loads; L2 hit).
    const float tz = th_rz[0];
    const float ty = th_ry[0];
    const float A  = __cosf(ty);               // cos(theta_ry)
    const float Bc = __sinf(ty) * __cosf(tz);  // sin(theta_ry)*cos(theta_rz)

    const int  tid  = (int)threadIdx.x;
    const long long base = (long long)blockIdx.x * (BLOCK * ITERS);
    // Wave-uniform first-lane id of this wave within the block.
    const int  waveBase = tid & ~31;

#if USE_ASYNC_LDS
    // Double-buffered async staging: each lane owns one 16B slot per buffer,
    // so no cross-lane sharing -> no workgroup barrier needed, only per-wave
    // s_wait_asynccnt ordering (async loads complete in issue order).
    __shared__ float4 buf[2][BLOCK];
    l_v4i* lbuf = (l_v4i*)&buf[0][0];
    g_v4i* gx   = (g_v4i*)x4;

    // Prologue: issue tile 0.
    {
        long long i0 = base + tid;
        if (i0 < n4) {
            __builtin_amdgcn_global_load_async_to_lds_b128(
                gx + i0, lbuf + tid, /*offset=*/0, /*cpol=*/0);
        }
    }

#pragma unroll
    for (int t = 0; t < ITERS; ++t) {
        long long inext0 = base + (long long)(t + 1) * BLOCK + waveBase;
        bool have_next = (t + 1 < ITERS) && (inext0 < (long long)n4);
        if (have_next) {
            long long inext = base + (long long)(t + 1) * BLOCK + tid;
            if (inext < n4) {
                __builtin_amdgcn_global_load_async_to_lds_b128(
                    gx + inext,
                    lbuf + ((t + 1) & 1) * BLOCK + tid,
                    /*offset=*/0, /*cpol=*/0);
            }
            // 2 outstanding -> ensure tile t landed, keep tile t+1 in flight.
            __builtin_amdgcn_s_wait_asynccnt(1);
        } else {
            __builtin_amdgcn_s_wait_asynccnt(0);
        }
        asm volatile("" ::: "memory");  // keep the ds_load after the wait

        long long i = base + (long long)t * BLOCK + tid;
        if (i < n4) {
            float4 v = buf[t & 1][tid];  // ds_load_b128 (own slot)
            float4 o;
            qmap(v.x, v.y, A, Bc, o.x, o.y);
            qmap(v.z, v.w, A, Bc, o.z, o.w);
            out4[i] = o;                 // global_store_b128
        }
    }
#else
    // Fallback: direct B128 streaming with explicit prefetch of the next tile
    // (global_prefetch_b8 on gfx1250).
#pragma unroll
    for (int t = 0; t < ITERS; ++t) {
        long long i = base + (long long)t * BLOCK + tid;
        if (t + 1 < ITERS) {
            long long inext = base + (long long)(t + 1) * BLOCK + tid;
            if (inext < n4) __builtin_prefetch(x4 + inext, 0, 1);
        }
        if (i < n4) {
            float4 v = x4[i];
            float4 o;
            qmap(v.x, v.y, A, Bc, o.x, o.y);
            qmap(v.z, v.w, A, Bc, o.z, o.w);
            out4[i] = o;
        }
    }
#endif
}

extern "C" void kernel_launch(void* const* d_in, const int* in_sizes, int n_in,
                              void* d_out, int out_size, void* d_ws, size_t ws_size,
                              hipStream_t stream) {
    (void)n_in; (void)d_ws; (void)ws_size; (void)out_size;

    const float4* x4    = (const float4*)d_in[0];   // x: [B,2] f32 -> B/2 float4
    const float*  th_rz = (const float*)d_in[1];
    const float*  th_ry = (const float*)d_in[2];
    float4*       out4  = (float4*)d_out;           // out: [B,2] f32

    const int n  = in_sizes[0];       // = B*2 = 8,388,608 (divisible by 4)
    const int n4 = n / 4;             // float4 chunks (2 batch elements each)

    const int perBlock = BLOCK * ITERS;
    const int grid = (n4 + perBlock - 1) / perBlock;   // 2048 blocks for B=4.19M

    qsim2_stream_kernel<<<grid, BLOCK, 0, stream>>>(x4, th_rz, th_ry, out4, n4);
}